// HashTable_72902774882594
// MI455X (gfx1250) — compile-verified
//
#include <hip/hip_runtime.h>
#include <hip/hip_bf16.h>
#include <math.h>

typedef __attribute__((ext_vector_type(2))) float v2f;
typedef __attribute__((ext_vector_type(8))) float v8f;

#define NPTS 256
#define MAXD2 2.0f  // MAX_DIST^2 = (sqrt2)^2 rounds to 2.0f

// Kernel 1: write pts output (256 x 3), one block.
__global__ __launch_bounds__(256) void pts_kernel(const float* __restrict__ sp,
                                                  float* __restrict__ out_pts) {
  int t = threadIdx.x;
  float theta = sp[2 * t + 0];
  float phi   = sp[2 * t + 1];
  float st = sinf(theta), ct = cosf(theta);
  float sph = sinf(phi),  cph = cosf(phi);
  out_pts[3 * t + 0] = cph * st;
  out_pts[3 * t + 1] = sph * st;
  out_pts[3 * t + 2] = ct;
}

// Kernel 2: block = (i, group of 4 j's). Computes inv(i,j) for the 4 j's,
// then uses V_WMMA_F32_16X16X4_F32 to produce hash[i,j,k,a] for all k,
// with rows m = 4*jj + a packing {4 j's} x {3 output components}.
__global__ __launch_bounds__(256) void hash_kernel(const float* __restrict__ sp,
                                                   float* __restrict__ out_hash,
                                                   float* __restrict__ out_mask) {
  const int i  = blockIdx.x;   // 0..255
  const int jg = blockIdx.y;   // 0..63 (j = 4*jg + jj)

  __shared__ float P[NPTS][4];      // pts, padded
  __shared__ float INVM[4][3][3];   // inv(i, jg*4+jj)
  __shared__ float VJ[4];           // valid(i,j) as 0/1
  __shared__ float VK[NPTS];        // valid(i,k) as 0/1

  const int t = threadIdx.x;

  // Recompute all 256 points into LDS (input is only 2 KB; cheaper than staging).
  {
    float theta = sp[2 * t + 0], phi = sp[2 * t + 1];
    float st = sinf(theta), ct = cosf(theta);
    float sph = sinf(phi),  cph = cosf(phi);
    P[t][0] = cph * st; P[t][1] = sph * st; P[t][2] = ct; P[t][3] = 0.f;
  }
  __syncthreads();

  const float pix = P[i][0], piy = P[i][1], piz = P[i][2];

  // valid(i, k) for every k
  {
    float dx = pix - P[t][0], dy = piy - P[t][1], dz = piz - P[t][2];
    float d2 = dx * dx + dy * dy + dz * dz;
    VK[t] = (d2 <= MAXD2 && t != i) ? 1.0f : 0.0f;
  }

  // 3x3 inverses for the 4 j's of this block (identity when invalid, as in ref,
  // so no Inf/NaN leaks into the masked multiply).
  if (t < 4) {
    int j = jg * 4 + t;
    float jx = P[j][0], jy = P[j][1], jz = P[j][2];
    float dx = pix - jx, dy = piy - jy, dz = piz - jz;
    bool valid = (dx * dx + dy * dy + dz * dz <= MAXD2) && (j != i);
    VJ[t] = valid ? 1.0f : 0.0f;

    float m00, m01, m02, m10, m11, m12, m20, m21, m22;
    if (valid) {
      float cx = piy * jz - piz * jy;
      float cy = piz * jx - pix * jz;
      float cz = pix * jy - piy * jx;
      // columns: p_i, p_j, p_i x p_j
      m00 = pix; m01 = jx; m02 = cx;
      m10 = piy; m11 = jy; m12 = cy;
      m20 = piz; m21 = jz; m22 = cz;
    } else {
      m00 = 1.f; m01 = 0.f; m02 = 0.f;
      m10 = 0.f; m11 = 1.f; m12 = 0.f;
      m20 = 0.f; m21 = 0.f; m22 = 1.f;
    }
    float c00 = m11 * m22 - m12 * m21;
    float c01 = m12 * m20 - m10 * m22;
    float c02 = m10 * m21 - m11 * m20;
    float det = m00 * c00 + m01 * c01 + m02 * c02;
    float r = 1.0f / det;
    INVM[t][0][0] = c00 * r;
    INVM[t][0][1] = (m02 * m21 - m01 * m22) * r;
    INVM[t][0][2] = (m01 * m12 - m02 * m11) * r;
    INVM[t][1][0] = c01 * r;
    INVM[t][1][1] = (m00 * m22 - m02 * m20) * r;
    INVM[t][1][2] = (m02 * m10 - m00 * m12) * r;
    INVM[t][2][0] = c02 * r;
    INVM[t][2][1] = (m01 * m20 - m00 * m21) * r;
    INVM[t][2][2] = (m00 * m11 - m01 * m10) * r;
  }
  __syncthreads();

  const int wave = t >> 5;       // 8 waves (wave32)
  const int lane = t & 31;
  const int half = lane >> 4;    // 0: lanes 0-15, 1: lanes 16-31
  const int l16  = lane & 15;

  // A-matrix (16x4 f32, §7.12.2): lane<16 -> VGPR0=K0, VGPR1=K1; lane>=16 -> K2, K3.
  // Row m = l16 = 4*jj + a packs inv rows; a==3 and K==3 are zero padding.
  const int jjA = l16 >> 2, aA = l16 & 3;
  float a0 = (aA < 3) ? INVM[jjA][aA][0] : 0.f;
  float a1 = (aA < 3) ? INVM[jjA][aA][1] : 0.f;
  float a2c = (aA < 3) ? INVM[jjA][aA][2] : 0.f;
  v2f Areg;
  Areg.x = (half == 0) ? a0 : a2c;  // K=0 / K=2
  Areg.y = (half == 0) ? a1 : 0.f;  // K=1 / K=3 (pad)

  // Each wave does 2 k-tiles of 16 -> k fully covered by 8 waves.
  for (int s = 0; s < 2; ++s) {
    const int k0 = (wave * 2 + s) * 16;
    const int kb = k0 + l16;

    // B-matrix (4x16 f32): row-striped across lanes; halves hold K and K+2.
    v2f Breg;
    Breg.x = (half == 0) ? P[kb][0] : P[kb][2];  // K=0 / K=2 rows
    Breg.y = (half == 0) ? P[kb][1] : 0.f;       // K=1 / K=3 (pad)

    v8f Creg = {0.f, 0.f, 0.f, 0.f, 0.f, 0.f, 0.f, 0.f};
    // 8 args: (neg_a, A, neg_b, B, c_mod, C, reuse_a, reuse_b)
    Creg = __builtin_amdgcn_wmma_f32_16x16x4_f32(
        false, Areg, false, Breg, (short)0, Creg, false, false);

    // C layout: VGPR r -> M = r + 8*half, N = l16. Fold mask into the store.
#pragma unroll
    for (int r2 = 0; r2 < 8; ++r2) {
      int mm = r2 + half * 8;
      int jj = mm >> 2, a = mm & 3;
      if (a < 3) {
        int j = jg * 4 + jj;
        float mv = VJ[jj] * VK[kb] * ((j != kb) ? 1.0f : 0.0f);
        size_t idx = (((size_t)i * NPTS + j) * NPTS + kb) * 3 + a;
        out_hash[idx] = Creg[r2] * mv;
      }
    }
  }

  // mask[i,j,k] plane: thread t handles k=t, coalesced across the block.
  {
    float vk = VK[t];
#pragma unroll
    for (int jj = 0; jj < 4; ++jj) {
      int j = jg * 4 + jj;
      out_mask[((size_t)i * NPTS + j) * NPTS + t] =
          VJ[jj] * vk * ((j != t) ? 1.0f : 0.0f);
    }
  }
}

extern "C" void kernel_launch(void* const* d_in, const int* in_sizes, int n_in,
                              void* d_out, int out_size, void* d_ws, size_t ws_size,
                              hipStream_t stream) {
  (void)in_sizes; (void)n_in; (void)d_ws; (void)ws_size; (void)out_size;
  const float* sp = (const float*)d_in[0];
  float* out = (float*)d_out;
  float* out_pts  = out;                                       // 256*3
  float* out_hash = out + 768;                                 // 256^3 * 3
  float* out_mask = out + 768 + (size_t)NPTS * NPTS * NPTS * 3; // 256^3

  pts_kernel<<<1, 256, 0, stream>>>(sp, out_pts);
  dim3 grid(NPTS, NPTS / 4);
  hash_kernel<<<grid, 256, 0, stream>>>(sp, out_hash, out_mask);
}